// LocalAttention_77275051590107
// MI455X (gfx1250) — compile-verified
//
#include <hip/hip_runtime.h>

typedef float v2f __attribute__((ext_vector_type(2)));
typedef float v8f __attribute__((ext_vector_type(8)));

#define B_DIM 32
#define S_DIM 2048
#define H_DIM 1024
#define ROWS_PER_WG 64
#define WIN 15   // exp(-d*d/2) == 0.0f in fp32 for |d| >= 15

// ---------------------------------------------------------------------------
// Kernel 1: q[32,1024] = dec[32,1024] @ Wa_w[1024,1024]  (fp32 WMMA 16x16x4)
// One wave per 16x16 output tile. EXEC all-ones (32 threads, no divergence).
// A (16x4 f32): lane<16 -> M=lane, {K=k+0,k+1}; lane>=16 -> M=lane-16, {K=k+2,k+3}
// B (4x16 f32): lane<16 -> N=lane, {K=k+0,k+1}; lane>=16 -> N=lane-16, {K=k+2,k+3}
// D (16x16 f32): VGPR v -> row m0 + 8*(lane>=16) + v, col n0 + (lane&15)
// ---------------------------------------------------------------------------
__global__ __launch_bounds__(32) void qgemm_wmma(const float* __restrict__ dec,
                                                 const float* __restrict__ Wa,
                                                 float* __restrict__ q) {
  const int n0   = blockIdx.x * 16;
  const int m0   = blockIdx.y * 16;
  const int l    = threadIdx.x;
  const int half = l >> 4;       // 0 or 1
  const int lm   = l & 15;

  const float* aRow = dec + (size_t)(m0 + lm) * H_DIM;  // A row (M = lm per half)
  const float* bCol = Wa + (n0 + lm);                   // B column (N = lm per half)

  v8f c = {};
  for (int k = 0; k < H_DIM; k += 4) {
    const int ka = k + 2 * half;
    v2f a, b;
    a.x = aRow[ka];
    a.y = aRow[ka + 1];
    b.x = bCol[(size_t)ka * H_DIM];
    b.y = bCol[(size_t)(ka + 1) * H_DIM];
    c = __builtin_amdgcn_wmma_f32_16x16x4_f32(false, a, false, b,
                                              (short)0, c, false, false);
  }

  const int row = m0 + half * 8;
#pragma unroll
  for (int v = 0; v < 8; ++v)
    q[(size_t)(row + v) * H_DIM + (n0 + lm)] = c[v];
}

// ---------------------------------------------------------------------------
// Kernel 2: scores[b,s] = q[b] . enc[b,s]   (streams 256 MB of enc once)
// 256 threads = 8 waves; each wave handles one row per iteration, 8 iters.
// ---------------------------------------------------------------------------
__global__ __launch_bounds__(256) void scores_kernel(const float* __restrict__ enc,
                                                     const float* __restrict__ q,
                                                     float* __restrict__ scores) {
  __shared__ float sq[H_DIM];
  const int chunks = S_DIM / ROWS_PER_WG;   // 32
  const int b      = blockIdx.x / chunks;
  const int chunk  = blockIdx.x % chunks;
  const int tid    = threadIdx.x;

  // stage q[b] (4 KB) into LDS
  ((float4*)sq)[tid] = ((const float4*)(q + (size_t)b * H_DIM))[tid];
  __syncthreads();

  const int wave = tid >> 5;
  const int lane = tid & 31;
  const size_t encB = (size_t)b * S_DIM * H_DIM;
  const float4* qv = (const float4*)sq;

  for (int i = 0; i < ROWS_PER_WG / 8; ++i) {
    const int s = chunk * ROWS_PER_WG + i * 8 + wave;
    const float4* rv = (const float4*)(enc + encB + (size_t)s * H_DIM);
    float acc = 0.f;
#pragma unroll
    for (int j = 0; j < 8; ++j) {
      const int idx = j * 32 + lane;     // 256 float4 = 1024 floats per row
      float4 e = rv[idx];
      float4 qq = qv[idx];
      acc += e.x * qq.x + e.y * qq.y + e.z * qq.z + e.w * qq.w;
    }
    // wave32 reduction
#pragma unroll
    for (int off = 16; off > 0; off >>= 1)
      acc += __shfl_xor(acc, off, 32);
    if (lane == 0) scores[b * S_DIM + s] = acc;
  }
}

// ---------------------------------------------------------------------------
// Kernel 3: per-batch softmax * gaussian -> attn; windowed context.
// One block (256 threads) per batch element.
// ---------------------------------------------------------------------------
__global__ __launch_bounds__(256) void finalize_kernel(const float* __restrict__ enc,
                                                       const float* __restrict__ scores,
                                                       const int* __restrict__ timestep,
                                                       float* __restrict__ ctx_out,
                                                       float* __restrict__ attn_out) {
  __shared__ float sAttn[S_DIM];   // 8 KB
  __shared__ float red[256];

  const int b   = blockIdx.x;
  const int tid = threadIdx.x;
  const int pt_i = timestep[b];
  const float pt = (float)pt_i;

  // load this batch's 2048 scores (8 per thread), local max
  float loc[8];
  float lmax = -__builtin_inff();
#pragma unroll
  for (int j = 0; j < 8; ++j) {
    loc[j] = scores[b * S_DIM + tid + j * 256];
    lmax = fmaxf(lmax, loc[j]);
  }
  red[tid] = lmax;
  __syncthreads();
  for (int off = 128; off > 0; off >>= 1) {
    if (tid < off) red[tid] = fmaxf(red[tid], red[tid + off]);
    __syncthreads();
  }
  const float smax = red[0];
  __syncthreads();

  // sum of exp(score - max)
  float ex[8];
  float lsum = 0.f;
#pragma unroll
  for (int j = 0; j < 8; ++j) {
    ex[j] = expf(loc[j] - smax);
    lsum += ex[j];
  }
  red[tid] = lsum;
  __syncthreads();
  for (int off = 128; off > 0; off >>= 1) {
    if (tid < off) red[tid] += red[tid + off];
    __syncthreads();
  }
  const float inv = 1.0f / red[0];
  __syncthreads();

  // attn = softmax * gaussian  (gaussian underflows to exact 0 for |d|>=15)
#pragma unroll
  for (int j = 0; j < 8; ++j) {
    const int s = tid + j * 256;
    const float d = (float)s - pt;
    const float g = expf(-0.5f * d * d);
    const float a = ex[j] * inv * g;
    sAttn[s] = a;
    attn_out[(size_t)b * S_DIM + s] = a;
  }
  __syncthreads();

  // context over the nonzero window only (exact: all other attn are 0.0f)
  const int s0 = (pt_i - WIN) < 0 ? 0 : (pt_i - WIN);
  const int s1 = (pt_i + WIN) > (S_DIM - 1) ? (S_DIM - 1) : (pt_i + WIN);
  const size_t encB = (size_t)b * S_DIM * H_DIM;
  float4 acc = {0.f, 0.f, 0.f, 0.f};
  for (int s = s0; s <= s1; ++s) {
    const float a = sAttn[s];
    float4 e = ((const float4*)(enc + encB + (size_t)s * H_DIM))[tid];
    acc.x += a * e.x;
    acc.y += a * e.y;
    acc.z += a * e.z;
    acc.w += a * e.w;
  }
  ((float4*)(ctx_out + (size_t)b * H_DIM))[tid] = acc;
}

// ---------------------------------------------------------------------------
extern "C" void kernel_launch(void* const* d_in, const int* in_sizes, int n_in,
                              void* d_out, int out_size, void* d_ws, size_t ws_size,
                              hipStream_t stream) {
  const float* enc = (const float*)d_in[0];   // [B,S,H]
  const float* dec = (const float*)d_in[1];   // [B,H]
  const int*   ts  = (const int*)d_in[2];     // [B]
  const float* Wa  = (const float*)d_in[3];   // [H,H]
  // d_in[4] = Wa_b: adds a per-batch constant to all scores -> softmax-invariant; skipped.

  float* out      = (float*)d_out;
  float* ctx_out  = out;                        // B*H floats
  float* attn_out = out + B_DIM * H_DIM;        // B*S floats

  float* q      = (float*)d_ws;                 // B*H floats
  float* scores = q + B_DIM * H_DIM;            // B*S floats

  qgemm_wmma<<<dim3(H_DIM / 16, B_DIM / 16), 32, 0, stream>>>(dec, Wa, q);
  scores_kernel<<<B_DIM * (S_DIM / ROWS_PER_WG), 256, 0, stream>>>(enc, q, scores);
  finalize_kernel<<<B_DIM, 256, 0, stream>>>(enc, scores, ts, ctx_out, attn_out);
}